// TestConv2_18322330484757
// MI455X (gfx1250) — compile-verified
//
#include <hip/hip_runtime.h>
#include <hip/hip_bf16.h>

#define N_MIRNA 812

typedef __attribute__((ext_vector_type(2))) float v2f;
typedef __attribute__((ext_vector_type(8))) float v8f;

static __device__ __forceinline__ v8f wmma_f32(v2f a, v2f b, v8f c) {
  // D = A(16x4 f32) * B(4x16 f32) + C(16x16 f32)
  return __builtin_amdgcn_wmma_f32_16x16x4_f32(
      /*neg_a=*/false, a, /*neg_b=*/false, b,
      /*c_mod=*/(short)0, c, /*reuse_a=*/false, /*reuse_b=*/false);
}

static __device__ __forceinline__ void atomAddF(float* p, float v) {
  __hip_atomic_fetch_add(p, v, __ATOMIC_RELAXED, __HIP_MEMORY_SCOPE_AGENT);
}

// ---------------------------------------------------------------------------
// Dense projection: ys = x @ W_same^T, yd = x @ W_diff^T   ([N,128]@[128,32])
// One wave per 16-row strip; 4 f32-WMMA accumulators (same/diff x two N-tiles)
// sharing one A tile per K-step. N is an exact multiple of 16 (50000 = 3125*16).
// ---------------------------------------------------------------------------
__global__ void gemm_pre_kernel(const float* __restrict__ x,
                                const float* __restrict__ Wsame,
                                const float* __restrict__ Wdiff,
                                float* __restrict__ ys,
                                float* __restrict__ yd, int N) {
  const int lane = threadIdx.x & 31;
  const int wave = threadIdx.x >> 5;
  const int tile = blockIdx.x * 4 + wave;
  const int m0 = tile * 16;
  if (m0 >= N) return;               // wave-uniform: EXEC stays all-1s for WMMA
  const int half = lane >> 4;        // 0: lanes 0-15, 1: lanes 16-31
  const int lm = lane & 15;

  v8f aS0 = {}, aS1 = {}, aD0 = {}, aD1 = {};
  const float* xrow = x + (size_t)(m0 + lm) * 128;   // A: row M=lm per lane
  const float* ws0 = Wsame + (size_t)lm * 128;        // B tile n0=0 : N=lm
  const float* ws1 = Wsame + (size_t)(16 + lm) * 128; // B tile n0=16
  const float* wd0 = Wdiff + (size_t)lm * 128;
  const float* wd1 = Wdiff + (size_t)(16 + lm) * 128;

  for (int k = 0; k < 128; k += 4) {
    // A 16x4 layout: v0 = K {k, k+2}, v1 = K {k+1, k+3} split by lane half
    const int kk = k + 2 * half;
    v2f a;  a.x = xrow[kk]; a.y = xrow[kk + 1];
    v2f b0; b0.x = ws0[kk]; b0.y = ws0[kk + 1];
    v2f b1; b1.x = ws1[kk]; b1.y = ws1[kk + 1];
    v2f c0; c0.x = wd0[kk]; c0.y = wd0[kk + 1];
    v2f c1; c1.x = wd1[kk]; c1.y = wd1[kk + 1];
    aS0 = wmma_f32(a, b0, aS0);
    aS1 = wmma_f32(a, b1, aS1);
    aD0 = wmma_f32(a, c0, aD0);
    aD1 = wmma_f32(a, c1, aD1);
  }
  // C/D layout: VGPR r -> M = r (lanes 0-15) / r+8 (lanes 16-31), N = lane&15
#pragma unroll
  for (int r = 0; r < 8; ++r) {
    const int row = m0 + r + 8 * half;
    ys[(size_t)row * 32 + lm]      = aS0[r];
    ys[(size_t)row * 32 + 16 + lm] = aS1[r];
    yd[(size_t)row * 32 + lm]      = aD0[r];
    yd[(size_t)row * 32 + 16 + lm] = aD1[r];
  }
}

// deg[c] += 1 for every edge target (matches torch_geometric degree over col)
__global__ void deg_kernel(const long long* __restrict__ ei, long long E,
                           float* __restrict__ deg) {
  long long stride = (long long)gridDim.x * blockDim.x;
  for (long long i = (long long)blockIdx.x * blockDim.x + threadIdx.x; i < E; i += stride) {
    atomAddF(&deg[(int)ei[E + i]], 1.0f);
  }
}

__global__ void dinv_kernel(const float* __restrict__ deg,
                            float* __restrict__ dinv, int N) {
  int i = blockIdx.x * blockDim.x + threadIdx.x;
  if (i < N) dinv[i] = rsqrtf(deg[i]);   // deg^-0.5 (inf for 0, same as ref)
}

// One wave per edge, lane == channel (wave32 == Cout). Each edge belongs to
// exactly one of {same, diff}; scatter norm * y_pre[col] into agg[row].
__global__ void agg_kernel(const long long* __restrict__ ei, long long E,
                           const float* __restrict__ dinv,
                           const float* __restrict__ ys,
                           const float* __restrict__ yd,
                           float* __restrict__ aggS,
                           float* __restrict__ aggD) {
  const int lane = threadIdx.x & 31;
  long long wid = ((long long)blockIdx.x * blockDim.x + threadIdx.x) >> 5;
  long long nw  = ((long long)gridDim.x * blockDim.x) >> 5;
  for (long long e = wid; e < E; e += nw) {
    const int r = (int)ei[e];
    const int c = (int)ei[E + e];
    const float nrm = dinv[r] * dinv[c];
    const bool same = (r < N_MIRNA) == (c < N_MIRNA);
    const float* src = same ? ys : yd;
    float* dst = same ? aggS : aggD;
    atomAddF(&dst[(size_t)r * 32 + lane], nrm * src[(size_t)c * 32 + lane]);
  }
}

__global__ void bias_diff_kernel(float* __restrict__ xd,
                                 const float* __restrict__ b, long long n32) {
  long long i = (long long)blockIdx.x * blockDim.x + threadIdx.x;
  if (i < n32) xd[i] += b[i & 31];
}

// Edge bilinear pooling over miRNA->disease edges only. W_bil staged in LDS
// with [i*32+j][o] layout (lane==o -> consecutive addresses, conflict-free).
// f1/f2 broadcast across the wave via shuffles.
__global__ void bilinear_kernel(const long long* __restrict__ ei, long long E,
                                const float* __restrict__ xdt,
                                const float* __restrict__ Wbil,
                                const float* __restrict__ bbil,
                                float* __restrict__ node_bil,
                                float* __restrict__ counts) {
  extern __shared__ float wsh[];  // 32*32*32 floats = 128 KB (of 320 KB WGP LDS)
  for (int t = threadIdx.x; t < 32 * 32 * 32; t += blockDim.x) {
    const int o = t >> 10;        // Wbil is [o][i][j] row-major in global
    const int ij = t & 1023;
    wsh[ij * 32 + o] = Wbil[t];
  }
  __syncthreads();

  const int lane = threadIdx.x & 31;   // lane == output channel o
  const float bo = bbil[lane];
  long long wid = ((long long)blockIdx.x * blockDim.x + threadIdx.x) >> 5;
  long long nw  = ((long long)gridDim.x * blockDim.x) >> 5;

  for (long long e = wid; e < E; e += nw) {
    const int r = (int)ei[e];
    const int c = (int)ei[E + e];
    if (!(r < N_MIRNA && c >= N_MIRNA)) continue;  // wave-uniform skip

    const float myf1 = xdt[(size_t)r * 32 + lane];
    const float myf2 = xdt[(size_t)c * 32 + lane];
    float f2r[32];
#pragma unroll
    for (int j = 0; j < 32; ++j) f2r[j] = __shfl(myf2, j);

    float acc = bo;
    for (int i = 0; i < 32; ++i) {
      const float f1i = __shfl(myf1, i);
      float partial = 0.0f;
      const float* wrow = &wsh[(i * 32) * 32 + lane];
#pragma unroll
      for (int j = 0; j < 32; ++j) partial += wrow[j * 32] * f2r[j];
      acc += f1i * partial;
    }
    atomAddF(&node_bil[(size_t)r * 32 + lane], acc);
    atomAddF(&node_bil[(size_t)c * 32 + lane], acc);
    if (lane == 0) {
      atomAddF(&counts[r], 1.0f);
      atomAddF(&counts[c], 1.0f);
    }
  }
}

__global__ void final_kernel(const float* __restrict__ aggS,
                             const float* __restrict__ bsame,
                             const float* __restrict__ xdt,
                             const float* __restrict__ node_bil,
                             const float* __restrict__ counts,
                             const float* __restrict__ gatew,
                             float* __restrict__ out, long long n32) {
  long long i = (long long)blockIdx.x * blockDim.x + threadIdx.x;
  if (i >= n32) return;
  const int o = (int)(i & 31);
  const long long n = i >> 5;
  const float g = 1.0f / (1.0f + __expf(-gatew[o]));
  const float bf = node_bil[i] / fmaxf(counts[n], 1.0f);
  const float lr = bf > 0.0f ? bf : 0.01f * bf;   // jax leaky_relu slope 0.01
  out[i] = (aggS[i] + bsame[o]) + g * lr + (1.0f - g) * xdt[i];
}

extern "C" void kernel_launch(void* const* d_in, const int* in_sizes, int n_in,
                              void* d_out, int out_size, void* d_ws, size_t ws_size,
                              hipStream_t stream) {
  const float*     x     = (const float*)d_in[0];
  const long long* ei    = (const long long*)d_in[1];
  const float*     Wsame = (const float*)d_in[2];
  const float*     bsame = (const float*)d_in[3];
  const float*     Wdiff = (const float*)d_in[4];
  const float*     bdiff = (const float*)d_in[5];
  const float*     Wbil  = (const float*)d_in[6];
  const float*     bbil  = (const float*)d_in[7];
  const float*     gatew = (const float*)d_in[8];

  const long long N = in_sizes[0] / 128;
  const long long E = in_sizes[1] / 2;

  float* ws    = (float*)d_ws;
  float* ysame = ws;                 // [N,32] projected features (same path)
  float* ydiff = ysame + N * 32;     // [N,32] projected features (diff path)
  float* dinv  = ydiff + N * 32;     // [N]
  float* aggS  = dinv + N;           // [N,32]  zeroed region starts here
  float* xdt   = aggS + N * 32;      // [N,32]  agg_diff, then +b_diff in place
  float* nbil  = xdt + N * 32;       // [N,32]
  float* counts = nbil + N * 32;     // [N]
  float* deg    = counts + N;        // [N]
  const size_t zbytes = (size_t)(3 * N * 32 + 2 * N) * sizeof(float);
  hipMemsetAsync(aggS, 0, zbytes, stream);

  const int tiles = (int)((N + 15) / 16);
  gemm_pre_kernel<<<(tiles + 3) / 4, 128, 0, stream>>>(x, Wsame, Wdiff, ysame, ydiff, (int)N);
  deg_kernel<<<1024, 256, 0, stream>>>(ei, E, deg);
  dinv_kernel<<<(int)((N + 255) / 256), 256, 0, stream>>>(deg, dinv, (int)N);
  agg_kernel<<<2048, 256, 0, stream>>>(ei, E, dinv, ysame, ydiff, aggS, xdt);
  bias_diff_kernel<<<(int)((N * 32 + 255) / 256), 256, 0, stream>>>(xdt, bdiff, N * 32);
  bilinear_kernel<<<512, 256, 32 * 32 * 32 * sizeof(float), stream>>>(
      ei, E, xdt, Wbil, bbil, nbil, counts);
  final_kernel<<<(int)((N * 32 + 255) / 256), 256, 0, stream>>>(
      aggS, bsame, xdt, nbil, counts, gatew, (float*)d_out, N * 32);
}